// FallacyAgent_67095979098190
// MI455X (gfx1250) — compile-verified
//
#include <hip/hip_runtime.h>
#include <math.h>

// ---------------------------------------------------------------------------
// EMA scan  g_t = (1-a)*g_{t-1} + a*u_t  over 2048 rows x 8192 timesteps.
// 3-pass chunked parallel scan; intra-chunk 16-step segments computed on the
// matrix core via V_WMMA_F32_16X16X4_F32 (lower-triangular coefficient matmul).
// Pass 3 double-buffers tiles into LDS with GLOBAL_LOAD_ASYNC_TO_LDS_B128
// (ASYNCcnt-tracked, no VGPR staging), fenced by s_wait_asynccnt/s_wait_dscnt.
// ---------------------------------------------------------------------------

typedef __attribute__((ext_vector_type(2))) float v2f;
typedef __attribute__((ext_vector_type(4))) float v4f;
typedef __attribute__((ext_vector_type(8))) float v8f;

#define ROWS   2048
#define TLEN   8192
#define CHUNKS 16
#define CLEN   (TLEN / CHUNKS)   // 512
#define RG     16                // rows per wave in pass 3
#define NBLK   (CLEN / 16)       // 32 16-step segments per chunk
#define SXS    36                // LDS row stride in floats (144B)

__device__ __forceinline__ float powu(float b, unsigned e) {
    float r = 1.0f, p = b;
    while (e) { if (e & 1u) r *= p; p *= p; e >>= 1u; }
    return r;
}

__device__ __forceinline__ float sigmoidf_dev(float v) {
    return 1.0f / (1.0f + __expf(-v));
}

// ---------------------------------------------------------------------------
// Pass 1: per-(row, chunk) exit state assuming zero entry state:
//   carry = alpha * sum_{j=0}^{L-1} a^{L-1-j} u_j
// Per-lane independent recurrence (acc = acc*a^32 + a^{31-lane}*u), then a
// wave32 shuffle reduction.  2048*16 = 32768 waves.
// ---------------------------------------------------------------------------
__global__ __launch_bounds__(256) void k_carry(const v2f* __restrict__ x,
                                               const float* __restrict__ a_raw,
                                               float* __restrict__ carry) {
    const float alpha = sigmoidf_dev(a_raw[0]);
    const float a     = 1.0f - alpha;

    const unsigned wid   = (blockIdx.x * blockDim.x + threadIdx.x) >> 5;
    const unsigned lane  = threadIdx.x & 31u;
    const unsigned row   = wid >> 4;          // / CHUNKS
    const unsigned chunk = wid & 15u;

    const float a32 = powu(a, 32u);
    const float wl  = powu(a, 31u - lane);

    const v2f* p = x + (size_t)row * TLEN + (size_t)chunk * CLEN + lane;

    float acc = 0.0f;
#pragma unroll
    for (int b = 0; b < CLEN / 32; ++b) {
        v2f v   = p[b * 32];
        float u = v.x - v.x * v.y;            // u = x0*(1-x1)
        acc = fmaf(acc, a32, wl * u);
    }
    for (int off = 16; off > 0; off >>= 1)
        acc += __shfl_down(acc, off, 32);
    if (lane == 0)
        carry[row * CHUNKS + chunk] = alpha * acc;
}

// ---------------------------------------------------------------------------
// Pass 2: per row, combine the 16 chunk carries sequentially and record each
// chunk's ENTRY state:  ginit[c] = G_{c-1};  G_c = carry_c + a^L * G_{c-1}.
// ---------------------------------------------------------------------------
__global__ __launch_bounds__(256) void k_combine(const float* __restrict__ carry,
                                                 const float* __restrict__ a_raw,
                                                 float* __restrict__ ginit) {
    const int row = blockIdx.x * blockDim.x + threadIdx.x;
    if (row >= ROWS) return;
    const float alpha = sigmoidf_dev(a_raw[0]);
    const float a     = 1.0f - alpha;
    const float aL    = powu(a, (unsigned)CLEN);
    float G = 0.0f;
#pragma unroll
    for (int c = 0; c < CHUNKS; ++c) {
        ginit[row * CHUNKS + c] = G;
        G = fmaf(aL, G, carry[row * CHUNKS + c]);
    }
}

// ---------------------------------------------------------------------------
// Pass 3: one wave per (16-row group, chunk).  For each 16-step segment:
//   out = S * U + p * g_in
// with S[i][j] = alpha*a^{i-j} (j<=i), p[i] = a^{i+1}.  S*U is 4 chained
// V_WMMA_F32_16X16X4_F32 (full f32 precision).  diff/delta recovered from
// consecutive outputs (delta = out_i - out_{i-1}, diff = delta/alpha).
// C/D layout: lane l, VGPR v  ->  M(time) = v + 8*(l>>4), N(row) = l&15.
// Tiles double-buffered in LDS via GLOBAL_LOAD_ASYNC_TO_LDS_B128.
// ---------------------------------------------------------------------------
__global__ __launch_bounds__(32) void k_scan(const float* __restrict__ x,
                                             const float* __restrict__ a_raw,
                                             const float* __restrict__ ginit,
                                             float* __restrict__ outp,
                                             float* __restrict__ diffp,
                                             float* __restrict__ deltap) {
    // raw x tile (16 rows x 16 timesteps x float2), double-buffered,
    // row stride 36 floats (144B: 16B-aligned async stores, spread banks)
    __shared__ __align__(16) float sX[2][16 * SXS];

    const float alpha     = sigmoidf_dev(a_raw[0]);
    const float a         = 1.0f - alpha;
    const float inv_alpha = 1.0f / alpha;

    const unsigned lane = threadIdx.x;
    const unsigned r    = lane & 15u;     // N = row within group
    const unsigned hi   = lane >> 4;      // lane-half: M offset 0 or 8

    const unsigned grp   = blockIdx.x >> 4;      // / CHUNKS
    const unsigned chunk = blockIdx.x & 15u;
    const unsigned row0  = grp * RG;
    const unsigned tbase = chunk * CLEN;

    // Constant A-matrix slices.  A layout (16x4 f32): lane -> M = lane&15,
    // VGPR{0,1} hold K = {0,1} (lanes 0-15) / {2,3} (lanes 16-31) of a slice.
    v2f A[4];
#pragma unroll
    for (int s = 0; s < 4; ++s) {
        const unsigned k0 = 4u * s + 2u * hi;
        A[s].x = (k0      <= r) ? alpha * powu(a, r - k0)        : 0.0f;
        A[s].y = (k0 + 1u <= r) ? alpha * powu(a, r - (k0 + 1u)) : 0.0f;
    }
    float P[8];
#pragma unroll
    for (int v = 0; v < 8; ++v)
        P[v] = powu(a, (unsigned)v + 8u * hi + 1u);

    // Cooperative tile-load mapping: lane pair (2k,2k+1) streams row k.
    const unsigned ldrow = lane >> 1;
    const unsigned half  = lane & 1u;
    const float* xrow = x + ((size_t)(row0 + ldrow) * TLEN + tbase + half * 8u) * 2u;

    // Per-lane LDS byte address of this lane's 64B staging slot (buffer 0).
    // Low 32 bits of a generic pointer into LDS == group-segment byte offset.
    const unsigned stageByte =
        (unsigned)(uintptr_t)&sX[0][0] + (ldrow * SXS + half * 16u) * 4u;
    const unsigned bufBytes = 16u * SXS * 4u;    // 2304

    // One b128 async copy moves 16B/lane; INST_OFFSET advances BOTH the
    // global and LDS addresses, so 4 issues with offset:0/16/32/48 move the
    // lane's whole 64B slice.
    auto issue_tile = [&](int j, int b) {
        const unsigned           dst = stageByte + (unsigned)b * bufBytes;
        const unsigned long long src = (unsigned long long)(uintptr_t)(xrow + j * 32);
        asm volatile(
            "global_load_async_to_lds_b128 %0, %1, off\n\t"
            "global_load_async_to_lds_b128 %0, %1, off offset:16\n\t"
            "global_load_async_to_lds_b128 %0, %1, off offset:32\n\t"
            "global_load_async_to_lds_b128 %0, %1, off offset:48"
            :: "v"(dst), "v"(src) : "memory");
    };

    issue_tile(0, 0);
    issue_tile(1, 1);

    float gin = ginit[(row0 + r) * CHUNKS + chunk];

    for (int blk = 0; blk < NBLK; ++blk) {
        const int bsel = blk & 1;
        const unsigned t0 = tbase + blk * 16u;

        // tile blk resident (its 4 copies retired; next tile's may be in flight)
        if (blk + 2 < NBLK)
            asm volatile("s_wait_asynccnt 0x4" ::: "memory");
        else
            asm volatile("s_wait_asynccnt 0x0" ::: "memory");

        // B slices: u[K][r] computed on read from raw (x0,x1) pairs in LDS
        v2f B[4];
#pragma unroll
        for (int s = 0; s < 4; ++s) {
            const unsigned kb = 4u * s + 2u * hi;
            const v2f p0 = *(const v2f*)&sX[bsel][r * SXS + 2u * kb];
            const v2f p1 = *(const v2f*)&sX[bsel][r * SXS + 2u * kb + 2u];
            B[s].x = p0.x - p0.x * p0.y;
            B[s].y = p1.x - p1.x * p1.y;
        }
        // LDS reads retired -> safe to overwrite this buffer
        asm volatile("s_wait_dscnt 0x0" ::: "memory");
        if (blk + 2 < NBLK) issue_tile(blk + 2, bsel);

        v8f c = {};
#pragma unroll
        for (int s = 0; s < 4; ++s)
            c = __builtin_amdgcn_wmma_f32_16x16x4_f32(
                    false, A[s], false, B[s], (short)0, c, false, false);

        // rank-1 fixup: out[i] = (S*U)[i] + a^{i+1} * g_in
        float o[8];
#pragma unroll
        for (int v = 0; v < 8; ++v) o[v] = c[v] + P[v] * gin;

        // g_prev for M = v+8*hi:  M==0 -> gin; M==8 -> other half's out[7]
        const float o7x = __shfl_xor(o[7], 16, 32);
        const float gp0 = hi ? o7x : gin;

        float dl[8], df[8];
        dl[0] = o[0] - gp0;
#pragma unroll
        for (int v = 1; v < 8; ++v) dl[v] = o[v] - o[v - 1];
#pragma unroll
        for (int v = 0; v < 8; ++v) df[v] = dl[v] * inv_alpha;

        // NT streamed stores: lane covers row0+r, t = t0 + 8*hi + [0..7]
        const size_t base = (size_t)(row0 + r) * TLEN + t0 + 8u * hi;
        v4f s0 = { o[0], o[1], o[2], o[3] };
        v4f s1 = { o[4], o[5], o[6], o[7] };
        __builtin_nontemporal_store(s0, (v4f*)(outp + base));
        __builtin_nontemporal_store(s1, (v4f*)(outp + base + 4));
        s0 = (v4f){ df[0], df[1], df[2], df[3] };
        s1 = (v4f){ df[4], df[5], df[6], df[7] };
        __builtin_nontemporal_store(s0, (v4f*)(diffp + base));
        __builtin_nontemporal_store(s1, (v4f*)(diffp + base + 4));
        s0 = (v4f){ dl[0], dl[1], dl[2], dl[3] };
        s1 = (v4f){ dl[4], dl[5], dl[6], dl[7] };
        __builtin_nontemporal_store(s0, (v4f*)(deltap + base));
        __builtin_nontemporal_store(s1, (v4f*)(deltap + base + 4));

        // carry: out[M=15] lives in hi-half VGPR 7; broadcast lane 16+r
        gin = __shfl(o[7], 16 + (int)r, 32);
    }
}

// ---------------------------------------------------------------------------
extern "C" void kernel_launch(void* const* d_in, const int* in_sizes, int n_in,
                              void* d_out, int out_size, void* d_ws, size_t ws_size,
                              hipStream_t stream) {
    (void)in_sizes; (void)n_in; (void)out_size; (void)ws_size;

    const v2f*   x     = (const v2f*)d_in[0];
    const float* a_raw = (const float*)d_in[1];

    float* outp   = (float*)d_out;
    float* diffp  = outp  + (size_t)ROWS * TLEN;
    float* deltap = diffp + (size_t)ROWS * TLEN;

    float* carry = (float*)d_ws;                 // ROWS*CHUNKS floats
    float* ginit = carry + ROWS * CHUNKS;        // ROWS*CHUNKS floats

    // Pass 1: 32768 waves (8 per block)
    k_carry<<<(ROWS * CHUNKS * 32) / 256, 256, 0, stream>>>(x, a_raw, carry);
    // Pass 2: one thread per row
    k_combine<<<ROWS / 256, 256, 0, stream>>>(carry, a_raw, ginit);
    // Pass 3: one wave per (16-row group, chunk) = 2048 waves
    k_scan<<<(ROWS / RG) * CHUNKS, 32, 0, stream>>>((const float*)x, a_raw, ginit,
                                                    outp, diffp, deltap);
}